// AudioPreprocessor_21792664060245
// MI455X (gfx1250) — compile-verified
//
#include <hip/hip_runtime.h>
#include <math.h>

typedef __attribute__((ext_vector_type(2))) float v2f;
typedef __attribute__((ext_vector_type(8))) float v8f;

namespace {
constexpr int   kNFFT  = 2048;
constexpr int   kHop   = 960;
constexpr int   kMels  = 64;
constexpr int   kDir   = 2304;
constexpr int   kB     = 8;
constexpr int   kSamp  = 960000;
constexpr int   kF     = 1025;            // NFFT/2+1
constexpr int   kKMain = 1024;            // guard-free K loop bound
constexpr int   kT     = 1001;            // frames
constexpr int   kFP    = 1028;            // K padded to multiple of 4
constexpr int   kTP    = 1008;            // N padded to multiple of 16
constexpr int   kTT    = kTP / 16;        // 63 t-tiles
constexpr float kEps   = 1e-8f;
constexpr size_t kArr  = (size_t)kB * kFP * kTP;   // one padded plane set
// workspace float offsets
constexpr size_t kPowL = 0;
constexpr size_t kPowR = kArr;
constexpr size_t kCsdR = 2 * kArr;
constexpr size_t kCsdI = 3 * kArr;
constexpr size_t kCnt  = 4 * kArr;
}

__device__ __forceinline__ v8f wmma_f32_k4(v2f a, v2f b, v8f c) {
  // D = A(16x4 f32) * B(4x16 f32) + C ; emits v_wmma_f32_16x16x4_f32
  return __builtin_amdgcn_wmma_f32_16x16x4_f32(false, a, false, b,
                                               (short)0, c, false, false);
}

__global__ void zero_f32(float* __restrict__ p, size_t n) {
  size_t i = (size_t)blockIdx.x * blockDim.x + threadIdx.x;
  if (i < n) p[i] = 0.f;
}

// One block per (t, b): windowed 2048-pt complex FFT holding L in real, R in
// imag; split spectra, emit pow_L/pow_R/csd_r/csd_i into padded [b][f][t].
__global__ __launch_bounds__(256) void stft_pow_csd(
    const float* __restrict__ wav, const float* __restrict__ win,
    float* __restrict__ ws) {
  __shared__ float zr[kNFFT];
  __shared__ float zi[kNFFT];
  const int t   = blockIdx.x;
  const int b   = blockIdx.y;
  const int tid = threadIdx.x;
  const float* wL = wav + (size_t)b * 2 * kSamp;
  const float* wR = wL + kSamp;
  const int base = t * kHop - kNFFT / 2;   // center padding

#pragma unroll
  for (int i = 0; i < kNFFT / 256; ++i) {
    int n = tid + i * 256;
    int s = base + n;
    bool ok = (s >= 0) && (s < kSamp);
    float w  = win[n];
    float xl = ok ? wL[s] * w : 0.f;
    float xr = ok ? wR[s] * w : 0.f;
    unsigned r = __brev((unsigned)n) >> 21;   // 11-bit reversal
    zr[r] = xl;
    zi[r] = xr;
  }
  __syncthreads();

  for (int len = 2; len <= kNFFT; len <<= 1) {
    int half = len >> 1;
    float ang0 = -6.28318530717958647692f / (float)len;
#pragma unroll
    for (int j = 0; j < (kNFFT / 2) / 256; ++j) {
      int idx = tid + j * 256;
      int blk = idx / half;
      int pos = idx - blk * half;
      int i0 = blk * len + pos;
      int i1 = i0 + half;
      float sn, cs;
      __sincosf(ang0 * (float)pos, &sn, &cs);
      float ar = zr[i1], ai = zi[i1];
      float tr = cs * ar - sn * ai;
      float ti = cs * ai + sn * ar;
      float br = zr[i0], bi = zi[i0];
      zr[i0] = br + tr; zi[i0] = bi + ti;
      zr[i1] = br - tr; zi[i1] = bi - ti;
    }
    __syncthreads();
  }

  for (int k = tid; k < kF; k += 256) {
    int k2 = (kNFFT - k) & (kNFFT - 1);
    float a_r = zr[k],  a_i = zi[k];
    float y_r = zr[k2], y_i = zi[k2];
    // X_L = (Z + conj(Zrev))/2 ; X_R = (Z - conj(Zrev))/(2i)
    float Lr = 0.5f * (a_r + y_r), Li = 0.5f * (a_i - y_i);
    float Rr = 0.5f * (a_i + y_i), Ri = 0.5f * (y_r - a_r);
    float pl = Lr * Lr + Li * Li;
    float pr = Rr * Rr + Ri * Ri;
    float cr = Lr * Rr + Li * Ri;   // Re(X_L * conj(X_R))
    float ci = Li * Rr - Lr * Ri;   // Im(X_L * conj(X_R))
    size_t o = ((size_t)b * kFP + k) * kTP + t;
    ws[kPowL + o] = pl;
    ws[kPowR + o] = pr;
    ws[kCsdR + o] = cr;
    ws[kCsdI + o] = ci;
  }
}

// One wave per 16x16 (m, t) tile: 4 WMMA f32 chains sharing the mel_fb
// A-fragment, fused log/sqrt pointwise into output channels 0..4.
// K-loop is guard-free (remainder handled branchlessly after the loop).
__global__ __launch_bounds__(32) void mel_channels(
    const float* __restrict__ melfb, const float* __restrict__ ws,
    float* __restrict__ out) {
  const int l   = threadIdx.x;
  const int t0  = blockIdx.x * 16;
  const int m0  = blockIdx.y * 16;
  const int b   = blockIdx.z;
  const int row = l & 15;          // A row (M)
  const int kh  = (l >> 4) * 2;    // K sub-offset per lane half
  const int col = l & 15;          // B/D column (N)
  const size_t bofs = (size_t)b * kFP * kTP;
  const float* arow = melfb + (size_t)(m0 + row) * kF;
  const float* bbase = ws + bofs + (size_t)(t0 + col);

  v8f accL = {}, accR = {}, accCr = {}, accCi = {};
  for (int k = 0; k < kKMain; k += 4) {
    int ka = k + kh;                       // ka+1 <= 1023 < kF: no guards
    v2f a;  a.x = arow[ka]; a.y = arow[ka + 1];
    size_t o = (size_t)ka * kTP;
    v2f bl, br2, bcr, bci;
    bl.x  = bbase[kPowL + o]; bl.y  = bbase[kPowL + o + kTP];
    br2.x = bbase[kPowR + o]; br2.y = bbase[kPowR + o + kTP];
    bcr.x = bbase[kCsdR + o]; bcr.y = bbase[kCsdR + o + kTP];
    bci.x = bbase[kCsdI + o]; bci.y = bbase[kCsdI + o + kTP];
    __builtin_prefetch(&bbase[kPowL + o + 4 * (size_t)kTP], 0, 1);
    __builtin_prefetch(&bbase[kCsdR + o + 4 * (size_t)kTP], 0, 1);
    accL  = wmma_f32_k4(a, bl,  accL);
    accR  = wmma_f32_k4(a, br2, accR);
    accCr = wmma_f32_k4(a, bcr, accCr);
    accCi = wmma_f32_k4(a, bci, accCi);
  }
  {
    // remainder f = 1024 (lanes with kh==2 contribute zero A; B pad rows are 0)
    float alast = arow[kKMain];            // valid address for every lane
    v2f a;  a.x = (kh == 0) ? alast : 0.f;  a.y = 0.f;
    int ka = kKMain + kh;
    size_t o = (size_t)ka * kTP;
    v2f bl, br2, bcr, bci;
    bl.x  = bbase[kPowL + o]; bl.y  = bbase[kPowL + o + kTP];
    br2.x = bbase[kPowR + o]; br2.y = bbase[kPowR + o + kTP];
    bcr.x = bbase[kCsdR + o]; bcr.y = bbase[kCsdR + o + kTP];
    bci.x = bbase[kCsdI + o]; bci.y = bbase[kCsdI + o + kTP];
    accL  = wmma_f32_k4(a, bl,  accL);
    accR  = wmma_f32_k4(a, br2, accR);
    accCr = wmma_f32_k4(a, bcr, accCr);
    accCi = wmma_f32_k4(a, bci, accCi);
  }

  int t = t0 + col;
  if (t < kT) {
    const size_t chs = (size_t)kMels * kT;
#pragma unroll
    for (int v = 0; v < 8; ++v) {
      int m = m0 + v + 8 * (l >> 4);
      float mL  = accL[v],  mR  = accR[v];
      float mcr = accCr[v], mci = accCi[v];
      float c0 = 10.f * log10f(mL + kEps);
      float c1 = 10.f * log10f(mR + kEps);
      float c2 = 10.f * log10f(mL / (mR + kEps) + kEps);
      float nrm = sqrtf(mcr * mcr + mci * mci + kEps);
      float c3 = mci / nrm;
      float c4 = mcr / nrm;
      size_t ob = (((size_t)b * 6 + 0) * kMels + m) * kT + t;
      out[ob]           = c0;
      out[ob + chs]     = c1;
      out[ob + 2 * chs] = c2;
      out[ob + 3 * chs] = c3;
      out[ob + 4 * chs] = c4;
    }
  }
}

// One wave per 32x16 (d, t) macro-tile: two d-tiles share the four streamed
// B fragments (8 WMMAs per K-step). Fused corr + atomic scatter into ch5.
__global__ __launch_bounds__(32) void corr_scatter(
    const float* __restrict__ Wr, const float* __restrict__ Wi,
    const float* __restrict__ Nr, const float* __restrict__ Nl,
    const int* __restrict__ az, const float* __restrict__ ws,
    float* __restrict__ out) {
  const int l   = threadIdx.x;
  const int t0  = blockIdx.x * 16;
  const int d0  = blockIdx.y * 32;
  const int b   = blockIdx.z;
  const int row = l & 15;
  const int kh  = (l >> 4) * 2;
  const int col = l & 15;
  const size_t bofs = (size_t)b * kFP * kTP;
  const size_t rb0  = (size_t)(d0 + row) * kF;        // first d-tile A row
  const size_t rb1  = (size_t)(d0 + 16 + row) * kF;   // second d-tile A row
  const float* bbase = ws + bofs + (size_t)(t0 + col);

  v8f aWr0 = {}, aWi0 = {}, aN10 = {}, aN20 = {};
  v8f aWr1 = {}, aWi1 = {}, aN11 = {}, aN21 = {};
  for (int k = 0; k < kKMain; k += 4) {
    int ka = k + kh;                       // ka+1 <= 1023 < kF: no guards
    v2f w0, i0, n10, n20, w1, i1, n11, n21;
    w0.x  = Wr[rb0 + ka]; w0.y  = Wr[rb0 + ka + 1];
    i0.x  = Wi[rb0 + ka]; i0.y  = Wi[rb0 + ka + 1];
    n10.x = Nr[rb0 + ka]; n10.y = Nr[rb0 + ka + 1];
    n20.x = Nl[rb0 + ka]; n20.y = Nl[rb0 + ka + 1];
    w1.x  = Wr[rb1 + ka]; w1.y  = Wr[rb1 + ka + 1];
    i1.x  = Wi[rb1 + ka]; i1.y  = Wi[rb1 + ka + 1];
    n11.x = Nr[rb1 + ka]; n11.y = Nr[rb1 + ka + 1];
    n21.x = Nl[rb1 + ka]; n21.y = Nl[rb1 + ka + 1];
    size_t o = (size_t)ka * kTP;
    v2f bcr, bci, bpl, bpr;
    bcr.x = bbase[kCsdR + o]; bcr.y = bbase[kCsdR + o + kTP];
    bci.x = bbase[kCsdI + o]; bci.y = bbase[kCsdI + o + kTP];
    bpl.x = bbase[kPowL + o]; bpl.y = bbase[kPowL + o + kTP];
    bpr.x = bbase[kPowR + o]; bpr.y = bbase[kPowR + o + kTP];
    __builtin_prefetch(&bbase[kCsdR + o + 4 * (size_t)kTP], 0, 1);
    __builtin_prefetch(&bbase[kPowL + o + 4 * (size_t)kTP], 0, 1);
    aWr0 = wmma_f32_k4(w0,  bcr, aWr0);
    aWi0 = wmma_f32_k4(i0,  bci, aWi0);
    aN10 = wmma_f32_k4(n10, bpl, aN10);
    aN20 = wmma_f32_k4(n20, bpr, aN20);
    aWr1 = wmma_f32_k4(w1,  bcr, aWr1);
    aWi1 = wmma_f32_k4(i1,  bci, aWi1);
    aN11 = wmma_f32_k4(n11, bpl, aN11);
    aN21 = wmma_f32_k4(n21, bpr, aN21);
  }
  {
    // remainder f = 1024, branchless (B pad rows >= 1025 are zero)
    float sel = (kh == 0) ? 1.f : 0.f;
    v2f w0, i0, n10, n20, w1, i1, n11, n21;
    w0.x  = sel * Wr[rb0 + kKMain]; w0.y  = 0.f;
    i0.x  = sel * Wi[rb0 + kKMain]; i0.y  = 0.f;
    n10.x = sel * Nr[rb0 + kKMain]; n10.y = 0.f;
    n20.x = sel * Nl[rb0 + kKMain]; n20.y = 0.f;
    w1.x  = sel * Wr[rb1 + kKMain]; w1.y  = 0.f;
    i1.x  = sel * Wi[rb1 + kKMain]; i1.y  = 0.f;
    n11.x = sel * Nr[rb1 + kKMain]; n11.y = 0.f;
    n21.x = sel * Nl[rb1 + kKMain]; n21.y = 0.f;
    int ka = kKMain + kh;
    size_t o = (size_t)ka * kTP;
    v2f bcr, bci, bpl, bpr;
    bcr.x = bbase[kCsdR + o]; bcr.y = bbase[kCsdR + o + kTP];
    bci.x = bbase[kCsdI + o]; bci.y = bbase[kCsdI + o + kTP];
    bpl.x = bbase[kPowL + o]; bpl.y = bbase[kPowL + o + kTP];
    bpr.x = bbase[kPowR + o]; bpr.y = bbase[kPowR + o + kTP];
    aWr0 = wmma_f32_k4(w0,  bcr, aWr0);
    aWi0 = wmma_f32_k4(i0,  bci, aWi0);
    aN10 = wmma_f32_k4(n10, bpl, aN10);
    aN20 = wmma_f32_k4(n20, bpr, aN20);
    aWr1 = wmma_f32_k4(w1,  bcr, aWr1);
    aWi1 = wmma_f32_k4(i1,  bci, aWi1);
    aN11 = wmma_f32_k4(n11, bpl, aN11);
    aN21 = wmma_f32_k4(n21, bpr, aN21);
  }

  int t = t0 + col;
  if (t < kT) {
    float* ch5 = out + (((size_t)b * 6 + 5) * kMels) * kT + t;
#pragma unroll
    for (int v = 0; v < 8; ++v) {
      int d = d0 + v + 8 * (l >> 4);
      float cu = aWr0[v] - aWi0[v];
      float dn = sqrtf(aN10[v] * aN20[v] + 1e-8f);
      atomicAdd(ch5 + (size_t)az[d] * kT, cu / dn);
    }
#pragma unroll
    for (int v = 0; v < 8; ++v) {
      int d = d0 + 16 + v + 8 * (l >> 4);
      float cu = aWr1[v] - aWi1[v];
      float dn = sqrtf(aN11[v] * aN21[v] + 1e-8f);
      atomicAdd(ch5 + (size_t)az[d] * kT, cu / dn);
    }
  }
}

__global__ void count_bins(const int* __restrict__ az, float* __restrict__ cnt) {
  int m = threadIdx.x;
  if (m < kMels) {
    float c = 0.f;
    for (int d = 0; d < kDir; ++d) c += (az[d] == m) ? 1.f : 0.f;
    cnt[m] = c;
  }
}

__global__ void norm_ch5(float* __restrict__ out, const float* __restrict__ cnt) {
  size_t i = (size_t)blockIdx.x * blockDim.x + threadIdx.x;
  size_t n = (size_t)kB * kMels * kT;
  if (i < n) {
    size_t b = i / ((size_t)kMels * kT);
    size_t m = (i / kT) % kMels;
    size_t t = i % kT;
    size_t o = (((size_t)b * 6 + 5) * kMels + m) * kT + t;
    out[o] = out[o] / (cnt[m] + kEps);
  }
}

extern "C" void kernel_launch(void* const* d_in, const int* in_sizes, int n_in,
                              void* d_out, int out_size, void* d_ws, size_t ws_size,
                              hipStream_t stream) {
  (void)in_sizes; (void)n_in; (void)ws_size;
  const float* wav   = (const float*)d_in[0];
  const float* Wr    = (const float*)d_in[1];
  const float* Wi    = (const float*)d_in[2];
  const float* Nr    = (const float*)d_in[3];
  const float* Nl    = (const float*)d_in[4];
  const int*   az    = (const int*)d_in[5];
  const float* win   = (const float*)d_in[6];
  const float* melfb = (const float*)d_in[7];
  float* out = (float*)d_out;
  float* ws  = (float*)d_ws;

  // Zero padded workspace planes (pad rows/cols must contribute 0 to GEMMs)
  {
    size_t n = 4 * kArr;
    zero_f32<<<dim3((unsigned)((n + 255) / 256)), dim3(256), 0, stream>>>(ws, n);
  }
  // Zero output (ch5 is accumulated via atomics)
  {
    size_t n = (size_t)out_size;
    zero_f32<<<dim3((unsigned)((n + 255) / 256)), dim3(256), 0, stream>>>(out, n);
  }

  stft_pow_csd<<<dim3(kT, kB), dim3(256), 0, stream>>>(wav, win, ws);
  mel_channels<<<dim3(kTT, kMels / 16, kB), dim3(32), 0, stream>>>(melfb, ws, out);
  corr_scatter<<<dim3(kTT, kDir / 32, kB), dim3(32), 0, stream>>>(Wr, Wi, Nr, Nl, az, ws, out);
  count_bins<<<dim3(1), dim3(64), 0, stream>>>(az, ws + kCnt);
  {
    size_t n = (size_t)kB * kMels * kT;
    norm_ch5<<<dim3((unsigned)((n + 255) / 256)), dim3(256), 0, stream>>>(out, ws + kCnt);
  }
}